// BuildK_70007966925527
// MI455X (gfx1250) — compile-verified
//
#include <hip/hip_runtime.h>
#include <stdint.h>

// Problem constants (fixed by the reference).
namespace {
constexpr int Hh   = 64;
constexpr int Nn   = 128;
constexpr int PTOT = Hh * Nn * Nn;        // 1,048,576 voxels
constexpr int KNB  = 9;                   // k nearest-in-intensity neighbors
constexpr int WROW = 12;                  // padded Wk row stride (48 B -> b128 aligned)
constexpr int TILE_ELEMS = 3 * 18 * 18;   // 972 halo slots per 16x16 tile
}

// ---------------------------------------------------------------------------
// Pass 1: per-voxel stable top-9 selection from the 3x3x3 window.
// Halo values staged into LDS with async global->LDS B32 copies (ASYNCcnt).
// Emits: Wk row (9 vals + pad) and the 9 selected *halo slot* indices.
// ---------------------------------------------------------------------------
__global__ __launch_bounds__(256) void nlm_select(const float* __restrict__ inp,
                                                  float* __restrict__ Wk,
                                                  int* __restrict__ IDl)
{
    __shared__ float tile[TILE_ELEMS];
    const int lx = threadIdx.x, ly = threadIdx.y;
    const int t  = ly * 16 + lx;
    const int x0 = blockIdx.x * 16, y0 = blockIdx.y * 16, z0 = blockIdx.z;

    // ---- async stage (z0-1..z0+1) x 18 x 18 halo values into LDS ----------
    const uint64_t base = (uint64_t)(uintptr_t)inp;
    #pragma unroll
    for (int it = 0; it < 4; ++it) {
        const int i = t + it * 256;
        if (i < TILE_ELEMS) {
            const int dz = i / 324;
            const int r  = i - dz * 324;
            const int dy = r / 18;
            const int dx = r - dy * 18;
            const int gz = (z0 + dz - 1 + Hh) & (Hh - 1);
            const int gy = (y0 + dy - 1 + Nn) & (Nn - 1);
            const int gx = (x0 + dx - 1 + Nn) & (Nn - 1);
            const unsigned goff  = (unsigned)((((gz * Nn) + gy) * Nn + gx) * 4);
            const unsigned laddr = (unsigned)(uintptr_t)(&tile[i]);
            asm volatile("global_load_async_to_lds_b32 %0, %1, %2"
                         :: "v"(laddr), "v"(goff), "s"(base) : "memory");
        }
    }
    asm volatile("s_wait_asynccnt 0" ::: "memory");
    __syncthreads();

    const int yy = ly + 1, xx = lx + 1;
    const float a = tile[324 + yy * 18 + xx];       // own value (dz=1 plane)

    // Stable top-9 (dist, halo-slot, value) kept in registers.
    float bd[KNB], bv[KNB];
    int   bi[KNB];
    #pragma unroll
    for (int i = 0; i < KNB; ++i) { bd[i] = 3.0e38f; bv[i] = 0.0f; bi[i] = 0; }

    // Candidate order matches the reference column order: oz, oy, ox.
    #pragma unroll
    for (int oz = -1; oz <= 1; ++oz)
    #pragma unroll
    for (int oy = -1; oy <= 1; ++oy)
    #pragma unroll
    for (int ox = -1; ox <= 1; ++ox) {
        const int slot = (oz + 1) * 324 + (yy + oy) * 18 + (xx + ox);
        const float v  = tile[slot];
        const float d  = fabsf(v - a);
        if (d < bd[KNB - 1]) {                      // strict '<' => stable ties
            bd[KNB - 1] = d; bv[KNB - 1] = v; bi[KNB - 1] = slot;
            #pragma unroll
            for (int i = KNB - 1; i > 0; --i) {
                if (bd[i] < bd[i - 1]) {            // bubble strictly past greater only
                    float td = bd[i]; bd[i] = bd[i - 1]; bd[i - 1] = td;
                    float tv = bv[i]; bv[i] = bv[i - 1]; bv[i - 1] = tv;
                    int   ti = bi[i]; bi[i] = bi[i - 1]; bi[i - 1] = ti;
                }
            }
        }
    }

    const int y = y0 + ly, x = x0 + lx;
    const int p = (z0 * Nn + y) * Nn + x;
    #pragma unroll
    for (int i = 0; i < KNB; ++i) {
        Wk[(size_t)p * WROW + i] = bv[i];
        IDl[(size_t)p * KNB + i] = bi[i];
    }
    Wk[(size_t)p * WROW + 9]  = 0.0f;   // pad -> rows store/load as 3x b128
    Wk[(size_t)p * WROW + 10] = 0.0f;
    Wk[(size_t)p * WROW + 11] = 0.0f;
}

// ---------------------------------------------------------------------------
// Pass 2: same tile mapping as pass 1. The 972 halo Wk rows (46.7 KB) are
// streamed into LDS with async B128 copies (one base pair, offset:0/16/32 —
// INST_OFFSET applies to both the LDS and global address per the ISA), then
// all neighbor-row gathers are ds_load_b128 from LDS. sigma is recomputed
// from the own row (cheaper than an 8 MB round trip). Softmax in registers.
// ---------------------------------------------------------------------------
__global__ __launch_bounds__(256) void nlm_dist_softmax(const float* __restrict__ Wk,
                                                        const int* __restrict__ IDl,
                                                        const float* __restrict__ ksig,
                                                        float* __restrict__ out)
{
    __shared__ float rows[TILE_ELEMS * WROW];       // 972 rows x 48 B = 46.7 KB
    const int lx = threadIdx.x, ly = threadIdx.y;
    const int t  = ly * 16 + lx;
    const int x0 = blockIdx.x * 16, y0 = blockIdx.y * 16, z0 = blockIdx.z;

    // ---- async stage halo Wk rows into LDS --------------------------------
    const uint64_t base = (uint64_t)(uintptr_t)Wk;
    #pragma unroll
    for (int it = 0; it < 4; ++it) {
        const int i = t + it * 256;
        if (i < TILE_ELEMS) {
            const int dz = i / 324;
            const int r  = i - dz * 324;
            const int dy = r / 18;
            const int dx = r - dy * 18;
            const int gz = (z0 + dz - 1 + Hh) & (Hh - 1);
            const int gy = (y0 + dy - 1 + Nn) & (Nn - 1);
            const int gx = (x0 + dx - 1 + Nn) & (Nn - 1);
            const int g  = ((gz * Nn) + gy) * Nn + gx;          // global voxel id
            const unsigned goff  = (unsigned)(g * (WROW * 4));  // row byte offset
            const unsigned laddr = (unsigned)(uintptr_t)(&rows[i * WROW]);
            asm volatile("global_load_async_to_lds_b128 %0, %1, %2"
                         :: "v"(laddr), "v"(goff), "s"(base) : "memory");
            asm volatile("global_load_async_to_lds_b128 %0, %1, %2 offset:16"
                         :: "v"(laddr), "v"(goff), "s"(base) : "memory");
            asm volatile("global_load_async_to_lds_b128 %0, %1, %2 offset:32"
                         :: "v"(laddr), "v"(goff), "s"(base) : "memory");
        }
    }
    asm volatile("s_wait_asynccnt 0" ::: "memory");
    __syncthreads();

    // ---- own row + sigma (ddof=1 std, same op order as pass-1 values) -----
    const int slot0 = 324 + (ly + 1) * 18 + (lx + 1);
    const float4* arow = (const float4*)(&rows[slot0 * WROW]);
    const float4 a0 = arow[0], a1 = arow[1], a2 = arow[2];
    const float a[KNB] = { a0.x, a0.y, a0.z, a0.w, a1.x, a1.y, a1.z, a1.w, a2.x };

    float mean = 0.0f;
    #pragma unroll
    for (int i = 0; i < KNB; ++i) mean += a[i];
    mean *= (1.0f / (float)KNB);
    float var = 0.0f;
    #pragma unroll
    for (int i = 0; i < KNB; ++i) { const float dv = a[i] - mean; var = fmaf(dv, dv, var); }
    const float sg = sqrtf(var * (1.0f / (float)(KNB - 1)));

    const float ks  = ksig[0];
    const float sgs = (sg == 0.0f) ? 1.0f : sg;
    const float scale = -1.0f / (2.0f * ks * ks * sgs * sgs);

    const int y = y0 + ly, x = x0 + lx;
    const int p = (z0 * Nn + y) * Nn + x;

    // ---- 9 neighbor rows from LDS, logits ---------------------------------
    float l[KNB];
    #pragma unroll
    for (int j = 0; j < KNB; ++j) {
        const int slot = IDl[(size_t)p * KNB + j];
        const float4* brow = (const float4*)(&rows[slot * WROW]);
        const float4 b0 = brow[0], b1 = brow[1], b2 = brow[2];
        const float b[KNB] = { b0.x, b0.y, b0.z, b0.w, b1.x, b1.y, b1.z, b1.w, b2.x };
        float acc = 0.0f;
        #pragma unroll
        for (int kk = 0; kk < KNB; ++kk) {
            const float d = a[kk] - b[kk] + 1e-6f;
            acc = fmaf(d, d, acc);
        }
        l[j] = (sg == 0.0f) ? 0.0f : acc * scale;   // logits use norm^2 directly
    }

    // ---- softmax over the 9 logits (max-subtracted) -----------------------
    float m = l[0];
    #pragma unroll
    for (int j = 1; j < KNB; ++j) m = fmaxf(m, l[j]);
    float e[KNB], s = 0.0f;
    #pragma unroll
    for (int j = 0; j < KNB; ++j) { e[j] = expf(l[j] - m); s += e[j]; }
    const float inv = 1.0f / s;
    #pragma unroll
    for (int j = 0; j < KNB; ++j) out[(size_t)p * KNB + j] = e[j] * inv;
}

// ---------------------------------------------------------------------------
// Launcher. Workspace layout (16B-aligned):
//   Wk  : PTOT * 12 floats  (48 MB)   -- padded rows, L2-resident in pass 2
//   IDl : PTOT * 9  ints    (36 MB)   -- selected halo-slot indices (0..971)
// ---------------------------------------------------------------------------
extern "C" void kernel_launch(void* const* d_in, const int* in_sizes, int n_in,
                              void* d_out, int out_size, void* d_ws, size_t ws_size,
                              hipStream_t stream)
{
    (void)in_sizes; (void)n_in; (void)out_size; (void)ws_size;

    const float* inp  = (const float*)d_in[0];   // (64,128,128) f32
    const float* ksig = (const float*)d_in[1];   // scalar f32

    float* Wk  = (float*)d_ws;
    int*   IDl = (int*)(Wk + (size_t)PTOT * WROW);

    dim3 blk(16, 16, 1), grd(Nn / 16, Nn / 16, Hh);
    nlm_select<<<grd, blk, 0, stream>>>(inp, Wk, IDl);
    nlm_dist_softmax<<<grd, blk, 0, stream>>>(Wk, IDl, ksig, (float*)d_out);
}